// Yuan_loss_22007412424754
// MI455X (gfx1250) — compile-verified
//
#include <hip/hip_runtime.h>
#include <stdint.h>

// Cross-entropy (sum) over [4, 2048, 32000] f32 logits, int64 targets.
// Memory-bound: 1.05 GB @ 23.3 TB/s => ~45 us floor. Strategy:
//   kernel 1: one workgroup per row; stream the row through LDS using
//             gfx1250 async global->LDS copies (double buffered, per-thread
//             private slots => no barriers in the hot loop, only
//             s_wait_asynccnt), online logsumexp in registers.
//   kernel 2: deterministic fixed-order reduction of 8192 row losses.

#define VOCAB    32000
#define ROW_V4   (VOCAB / 4)          // 8000 float4 per row
#define TPB      256                  // 8 wave32 per workgroup
#define WAVES    (TPB / 32)
#define CHUNK_V4 1792                 // float4 per chunk (28 KB)
#define ROUNDS   (CHUNK_V4 / TPB)     // 7 async issues per wave per chunk
#define NCHUNK   ((ROW_V4 + CHUNK_V4 - 1) / CHUNK_V4)   // 5

static_assert(ROUNDS * TPB == CHUNK_V4, "chunk must tile evenly");

#define NEG_HUGE (-3.402823466e38f)

__device__ __forceinline__ void combine_ms(float& m, float& s, float mo, float so) {
    float M = fmaxf(m, mo);
    s = s * __expf(m - M) + so * __expf(mo - M);
    m = M;
}

__global__ __launch_bounds__(TPB) void
ce_rowloss_kernel(const float* __restrict__ x, const long long* __restrict__ y,
                  float* __restrict__ row_loss) {
    __shared__ float4 buf[2][CHUNK_V4];        // 2 x 28 KB staging buffers
    __shared__ float red_m[WAVES];
    __shared__ float red_s[WAVES];

    const int tid = threadIdx.x;
    const int row = blockIdx.x;
    const float4* xr4 = (const float4*)(x + (size_t)row * VOCAB);

    // Issue one chunk of async global->LDS B128 copies. Every wave issues
    // exactly ROUNDS instructions (tail slots are clamped to a valid slot:
    // duplicate write of identical data -> benign), so the scalar
    // s_wait_asynccnt thresholds below are uniform per wave.
    auto issue_chunk = [&](int c, int b) {
        const int base = c * CHUNK_V4;
#pragma unroll
        for (int r = 0; r < ROUNDS; ++r) {
            int g = base + r * TPB + tid;
            g = (g < ROW_V4) ? g : (ROW_V4 - 1);
            const int es = g - base;                       // slot within chunk
            unsigned lds = (unsigned)(uintptr_t)(&buf[b][es]); // low 32 bits of the
                                                               // shared aperture = LDS offset
            const float4* gp = xr4 + g;
            asm volatile("global_load_async_to_lds_b128 %0, %1, off"
                         :: "v"(lds), "v"(gp) : "memory");
        }
    };

    float m = NEG_HUGE, s = 0.0f;

    issue_chunk(0, 0);
#pragma unroll
    for (int c = 0; c < NCHUNK; ++c) {
        const int b = c & 1;
        if (c + 1 < NCHUNK) {
            issue_chunk(c + 1, (c + 1) & 1);   // prefetch next buffer
            // 7 newest (next chunk) may remain outstanding; current chunk done.
            asm volatile("s_wait_asynccnt 0x7" ::: "memory");
        } else {
            asm volatile("s_wait_asynccnt 0x0" ::: "memory");
        }
        const int base = c * CHUNK_V4;
#pragma unroll
        for (int r = 0; r < ROUNDS; ++r) {
            const int slot = r * TPB + tid;
            if (base + slot < ROW_V4) {            // only last chunk masks
                float4 a = buf[b][slot];
                float v  = fmaxf(fmaxf(a.x, a.y), fmaxf(a.z, a.w));
                float M  = fmaxf(m, v);
                float sc = __expf(m - M);          // 1.0 when max unchanged
                s = s * sc + (__expf(a.x - M) + __expf(a.y - M))
                           + (__expf(a.z - M) + __expf(a.w - M));
                m = M;
            }
        }
    }

    // wave32 (m, s) reduction
#pragma unroll
    for (int off = 16; off > 0; off >>= 1) {
        float mo = __shfl_xor(m, off, 32);
        float so = __shfl_xor(s, off, 32);
        combine_ms(m, s, mo, so);
    }
    if ((tid & 31) == 0) { red_m[tid >> 5] = m; red_s[tid >> 5] = s; }
    __syncthreads();
    if (tid == 0) {
        float M = red_m[0], S = red_s[0];
#pragma unroll
        for (int w = 1; w < WAVES; ++w) combine_ms(M, S, red_m[w], red_s[w]);
        const int t  = (int)y[row];
        const float xt = x[(size_t)row * VOCAB + t];
        row_loss[row] = M + __logf(S) - xt;        // NLL of target token
    }
}

// Fixed-order, single-block reduction => deterministic across replays.
__global__ __launch_bounds__(256) void
ce_reduce_kernel(const float* __restrict__ row_loss, float* __restrict__ out, int n) {
    __shared__ float sh[256];
    const int t = threadIdx.x;
    float s = 0.0f;
    for (int i = t; i < n; i += 256) s += row_loss[i];
    sh[t] = s;
    __syncthreads();
#pragma unroll
    for (int off = 128; off > 0; off >>= 1) {
        if (t < off) sh[t] += sh[t + off];
        __syncthreads();
    }
    if (t == 0) out[0] = sh[0];
}

extern "C" void kernel_launch(void* const* d_in, const int* in_sizes, int n_in,
                              void* d_out, int out_size, void* d_ws, size_t ws_size,
                              hipStream_t stream) {
    const float*     x = (const float*)d_in[0];       // [4, 2048, 32000] f32
    const long long* y = (const long long*)d_in[1];   // [4, 2048] i64
    float* out      = (float*)d_out;                  // scalar f32
    float* row_loss = (float*)d_ws;                   // 8192 f32 scratch

    const int rows = in_sizes[1];                     // 4 * 2048 = 8192
    ce_rowloss_kernel<<<rows, TPB, 0, stream>>>(x, y, row_loss);
    ce_reduce_kernel<<<1, 256, 0, stream>>>(row_loss, out, rows);
}